// KANC_MLP_8306466751007
// MI455X (gfx1250) — compile-verified
//
#include <hip/hip_runtime.h>
#include <hip/hip_bf16.h>

typedef float v2f __attribute__((ext_vector_type(2)));
typedef float v8f __attribute__((ext_vector_type(8)));

// ---------------------------------------------------------------------------
// B-spline basis (GRID_SIZE=5, ORDER=3, knots = -2.2 + 0.4*i, i=0..11)
// Cox-de Boor; uniform knots so both denominators are j*h.
// ---------------------------------------------------------------------------
__device__ __forceinline__ void bspline8(float x, float bas[8]) {
    float b[11];
#pragma unroll
    for (int i = 0; i < 11; ++i) {
        float lo = -2.2f + 0.4f * (float)i;
        b[i] = (x >= lo && x < lo + 0.4f) ? 1.0f : 0.0f;
    }
#pragma unroll
    for (int j = 1; j <= 3; ++j) {
        float inv = 1.0f / (0.4f * (float)j);
#pragma unroll
        for (int i = 0; i + j < 11; ++i) {
            float lo = -2.2f + 0.4f * (float)i;
            float hi = -2.2f + 0.4f * (float)(i + j + 1);
            b[i] = (x - lo) * inv * b[i] + (hi - x) * inv * b[i + 1];
        }
    }
#pragma unroll
    for (int i = 0; i < 8; ++i) bas[i] = b[i];
}

__device__ __forceinline__ float siluf(float v) {
    return v / (1.0f + __expf(-v));
}

// ---------------------------------------------------------------------------
// Repack conv weights into WMMA-B layout: wt[n][kk], n=0..15 (zero for n>=OC),
// kk: [0,45) = base weights, [45,48) zero pad, [48,408) = spline weights.
// ---------------------------------------------------------------------------
__global__ void build_wt(const float* __restrict__ bw, const float* __restrict__ sw,
                         float* __restrict__ wt, int OC) {
    for (int e = threadIdx.x; e < 16 * 408; e += blockDim.x) {
        int n = e / 408, kk = e % 408;
        float v = 0.0f;
        if (n < OC) {
            if (kk < 45) v = bw[n * 45 + kk];
            else if (kk >= 48) {
                int i = (kk - 48) >> 3, k = (kk - 48) & 7;
                v = sw[(n * 45 + i) * 8 + k];
            }
        }
        wt[e] = v;
    }
}

// ---------------------------------------------------------------------------
// Layer 1: KAN conv (C=1, 3x3, out=5) fused with 2x2 maxpool. Pure VALU:
// in=9 is too small for the matrix pipe; the bspline recursion dominates.
// One lane per pooled output (b, ph, pw) -> 4 conv positions x 5 channels.
// ---------------------------------------------------------------------------
__launch_bounds__(128)
__global__ void kan_conv1_pool(const float* __restrict__ x, const float* __restrict__ bw,
                               const float* __restrict__ sw, float* __restrict__ h1,
                               int Npool) {
    __shared__ __align__(16) float wbL[48];
    __shared__ __align__(16) float wsL[360];
    for (int e = threadIdx.x; e < 45; e += blockDim.x) wbL[e] = bw[e];
    for (int e = threadIdx.x; e < 360; e += blockDim.x) wsL[e] = sw[e];
    __syncthreads();

    int t = blockIdx.x * blockDim.x + threadIdx.x;
    if (t >= Npool) return;
    int b = t / 169, r = t % 169;
    int ph = r / 13, pw = r % 13;
    const float* xb = x + b * 784 + (2 * ph) * 28 + 2 * pw;

    float patch[4][4];
#pragma unroll
    for (int rr = 0; rr < 4; ++rr)
#pragma unroll
        for (int cc = 0; cc < 4; ++cc)
            patch[rr][cc] = xb[rr * 28 + cc];

    float o[4][5];
#pragma unroll
    for (int p = 0; p < 4; ++p)
#pragma unroll
        for (int c = 0; c < 5; ++c) o[p][c] = 0.0f;

#pragma unroll
    for (int i = 0; i < 9; ++i) {
        const int dy = i / 3, dx = i % 3;
        float s4[4], bas4[4][8];
#pragma unroll
        for (int p = 0; p < 4; ++p) {
            float v = patch[(p >> 1) + dy][(p & 1) + dx];
            s4[p] = siluf(v);
            bspline8(v, bas4[p]);
        }
#pragma unroll
        for (int c = 0; c < 5; ++c) {
            float  w0  = wbL[c * 9 + i];
            float4 wk0 = *(const float4*)&wsL[(c * 9 + i) * 8];
            float4 wk1 = *(const float4*)&wsL[(c * 9 + i) * 8 + 4];
#pragma unroll
            for (int p = 0; p < 4; ++p) {
                o[p][c] += w0 * s4[p]
                         + wk0.x * bas4[p][0] + wk0.y * bas4[p][1]
                         + wk0.z * bas4[p][2] + wk0.w * bas4[p][3]
                         + wk1.x * bas4[p][4] + wk1.y * bas4[p][5]
                         + wk1.z * bas4[p][6] + wk1.w * bas4[p][7];
            }
        }
    }
#pragma unroll
    for (int c = 0; c < 5; ++c) {
        float m = fmaxf(fmaxf(o[0][c], o[1][c]), fmaxf(o[2][c], o[3][c]));
        h1[(b * 5 + c) * 169 + r] = m;
    }
}

// ---------------------------------------------------------------------------
// Layers 2/3: KAN conv via fp32 WMMA. One wave = 32 positions.
// Phase 1: each lane builds its feature row (45 silu + pad + 360 spline) in
//          LDS; row stride 420 floats -> bank-conflict-free b64/b128 access.
// Phase 2: 102 software-pipelined k-steps of v_wmma_f32_16x16x4_f32, two
//          16x16 M-tiles per wave. No branches in the loop; EXEC all-ones.
// ---------------------------------------------------------------------------
template <int CIN, int HI, int WI, int HO, int WO, int OC>
__launch_bounds__(32)
__global__ void kan_conv_wmma(const float* __restrict__ in, const float* __restrict__ wt,
                              float* __restrict__ out, int Npos) {
    constexpr int IN    = CIN * 9;        // 45
    constexpr int SOFF  = 48;
    constexpr int KK    = SOFF + IN * 8;  // 408 (real K extent)
    constexpr int FSTR  = 420;            // padded LDS stride (420 % 64 = 36)
    constexpr int NSTEP = KK / 4;         // 102
    __shared__ __align__(16) float feats[32 * FSTR];  // 52.5 KB

    const int lane = threadIdx.x;
    const int pos0 = blockIdx.x * 32;
    int pos = pos0 + lane;
    int cp  = (pos < Npos) ? pos : (Npos - 1);   // clamp; OOB masked at store
    int b   = cp / (HO * WO);
    int rr0 = cp % (HO * WO);
    int oh  = rr0 / WO, ow = rr0 % WO;

    float* f = &feats[lane * FSTR];
    f[IN] = 0.0f; f[IN + 1] = 0.0f; f[IN + 2] = 0.0f;  // pad 45..47
#pragma unroll
    for (int c = 0; c < CIN; ++c)
#pragma unroll
        for (int dy = 0; dy < 3; ++dy)
#pragma unroll
            for (int dx = 0; dx < 3; ++dx) {
                int i = (c * 3 + dy) * 3 + dx;
                float v = in[((b * CIN + c) * HI + (oh + dy)) * WI + (ow + dx)];
                f[i] = siluf(v);
                float bas[8];
                bspline8(v, bas);
                *(float4*)(&f[SOFF + i * 8])     = make_float4(bas[0], bas[1], bas[2], bas[3]);
                *(float4*)(&f[SOFF + i * 8 + 4]) = make_float4(bas[4], bas[5], bas[6], bas[7]);
            }
    __syncthreads();

    const int khalf = (lane >> 4) << 1;   // 0 or 2
    const int mrow  = lane & 15;
    const int n     = lane & 15;
    const float* f0 = &feats[mrow * FSTR + khalf];
    const float* f1 = &feats[(16 + mrow) * FSTR + khalf];
    const float* wr = wt + n * KK + khalf;

    v8f acc0 = {}; v8f acc1 = {};
    v2f a0 = *(const v2f*)f0;
    v2f a1 = *(const v2f*)f1;
    v2f bb = *(const v2f*)wr;
    for (int kk = 0; kk < NSTEP - 1; ++kk) {
        int nxt = (kk + 1) << 2;
        v2f a0n = *(const v2f*)(f0 + nxt);
        v2f a1n = *(const v2f*)(f1 + nxt);
        v2f bbn = *(const v2f*)(wr + nxt);
        acc0 = __builtin_amdgcn_wmma_f32_16x16x4_f32(false, a0, false, bb, (short)0, acc0, false, false);
        acc1 = __builtin_amdgcn_wmma_f32_16x16x4_f32(false, a1, false, bb, (short)0, acc1, false, false);
        a0 = a0n; a1 = a1n; bb = bbn;
    }
    acc0 = __builtin_amdgcn_wmma_f32_16x16x4_f32(false, a0, false, bb, (short)0, acc0, false, false);
    acc1 = __builtin_amdgcn_wmma_f32_16x16x4_f32(false, a1, false, bb, (short)0, acc1, false, false);

    if (n < OC) {
#pragma unroll
        for (int r = 0; r < 8; ++r) {
            int m  = r + ((lane >> 4) << 3);     // C/D layout: M = r (+8 hi half)
            int p0 = pos0 + m;
            int p1 = pos0 + 16 + m;
            if (p0 < Npos) {
                int bb2 = p0 / (HO * WO), r2 = p0 % (HO * WO);
                out[((bb2 * OC + n) * HO + (r2 / WO)) * WO + (r2 % WO)] = acc0[r];
            }
            if (p1 < Npos) {
                int bb2 = p1 / (HO * WO), r2 = p1 % (HO * WO);
                out[((bb2 * OC + n) * HO + (r2 / WO)) * WO + (r2 % WO)] = acc1[r];
            }
        }
    }
}

// ---------------------------------------------------------------------------
// fp32 WMMA GEMM: C[M,N] = A[M,K] * W[N,K]^T + bias (+ ReLU).
// One wave computes a 16 x (16*NTPW) macro-tile: A fragment reused NTPW times.
// Main k-loop is branch-free (row indices clamped; garbage confined to rows/
// cols that the store masks off). K%4==2 handled by one zero-padded epilogue.
// ---------------------------------------------------------------------------
template <int NTPW>
__launch_bounds__(256)
__global__ void fc_wmma(const float* __restrict__ A, const float* __restrict__ W,
                        const float* __restrict__ bias, float* __restrict__ C,
                        int M, int N, int K, int NG, int doRelu) {
    int wave = (blockIdx.x * blockDim.x + threadIdx.x) >> 5;
    int lane = threadIdx.x & 31;
    int mt = wave / NG, g = wave % NG;
    int khalf = (lane >> 4) << 1;

    int mrow = mt * 16 + (lane & 15);
    int mL   = (mrow < M) ? mrow : (M - 1);
    int nbase = g * (16 * NTPW) + (lane & 15);

    const float* Arow = A + (long)mL * K + khalf;
    const float* Brow[NTPW];
#pragma unroll
    for (int t = 0; t < NTPW; ++t) {
        int nt = nbase + 16 * t;
        int nL = (nt < N) ? nt : (N - 1);
        Brow[t] = W + (long)nL * K + khalf;
    }

    v8f acc[NTPW];
#pragma unroll
    for (int t = 0; t < NTPW; ++t) acc[t] = (v8f){};

    const int kmain = K >> 2;
    v2f a = *(const v2f*)Arow;
    v2f bv[NTPW];
#pragma unroll
    for (int t = 0; t < NTPW; ++t) bv[t] = *(const v2f*)Brow[t];

    for (int kk = 0; kk < kmain - 1; ++kk) {
        int nxt = (kk + 1) << 2;
        v2f an = *(const v2f*)(Arow + nxt);
        v2f bn[NTPW];
#pragma unroll
        for (int t = 0; t < NTPW; ++t) bn[t] = *(const v2f*)(Brow[t] + nxt);
#pragma unroll
        for (int t = 0; t < NTPW; ++t)
            acc[t] = __builtin_amdgcn_wmma_f32_16x16x4_f32(false, a, false, bv[t], (short)0, acc[t], false, false);
        a = an;
#pragma unroll
        for (int t = 0; t < NTPW; ++t) bv[t] = bn[t];
    }
#pragma unroll
    for (int t = 0; t < NTPW; ++t)
        acc[t] = __builtin_amdgcn_wmma_f32_16x16x4_f32(false, a, false, bv[t], (short)0, acc[t], false, false);

    if (K & 3) {  // K % 4 == 2 (K always even here): one zero-padded step
        v2f az = {0.0f, 0.0f};
        v2f bz[NTPW];
#pragma unroll
        for (int t = 0; t < NTPW; ++t) bz[t] = (v2f){0.0f, 0.0f};
        if (khalf == 0) {  // lanes 0-15 cover k = K-2, K-1; hi half stays zero
            az = *(const v2f*)(Arow + (kmain << 2));
#pragma unroll
            for (int t = 0; t < NTPW; ++t) bz[t] = *(const v2f*)(Brow[t] + (kmain << 2));
        }
#pragma unroll
        for (int t = 0; t < NTPW; ++t)
            acc[t] = __builtin_amdgcn_wmma_f32_16x16x4_f32(false, az, false, bz[t], (short)0, acc[t], false, false);
    }

#pragma unroll
    for (int t = 0; t < NTPW; ++t) {
        int nt = nbase + 16 * t;
        if (nt < N) {
            float bzv = bias[nt];
#pragma unroll
            for (int r = 0; r < 8; ++r) {
                int m = mt * 16 + r + ((lane >> 4) << 3);
                if (m < M) {
                    float v = acc[t][r] + bzv;
                    if (doRelu) v = fmaxf(v, 0.0f);
                    C[(long)m * N + nt] = v;
                }
            }
        }
    }
}

// ---------------------------------------------------------------------------
extern "C" void kernel_launch(void* const* d_in, const int* in_sizes, int n_in,
                              void* d_out, int out_size, void* d_ws, size_t ws_size,
                              hipStream_t stream) {
    (void)n_in; (void)out_size; (void)ws_size;
    const float* x   = (const float*)d_in[0];
    const float* bw1 = (const float*)d_in[1];
    const float* sw1 = (const float*)d_in[2];
    const float* bw2 = (const float*)d_in[3];
    const float* sw2 = (const float*)d_in[4];
    const float* bw3 = (const float*)d_in[5];
    const float* sw3 = (const float*)d_in[6];
    const float* w1  = (const float*)d_in[7];
    const float* b1  = (const float*)d_in[8];
    const float* w2  = (const float*)d_in[9];
    const float* b2  = (const float*)d_in[10];
    float* out = (float*)d_out;

    int B = in_sizes[0] / 784;

    // workspace layout (floats)
    float* wsf = (float*)d_ws;
    size_t o = 0;
    float* h1  = wsf + o; o += (size_t)B * 845;   // [B,5,13,13]
    float* h2  = wsf + o; o += (size_t)B * 605;   // [B,5,11,11]
    float* h3  = wsf + o; o += (size_t)B * 162;   // [B,162]
    float* fc1 = wsf + o; o += (size_t)B * 500;   // [B,500]
    float* wt2 = wsf + o; o += 16 * 408;
    float* wt3 = wsf + o; o += 16 * 408;

    build_wt<<<1, 256, 0, stream>>>(bw2, sw2, wt2, 5);
    build_wt<<<1, 256, 0, stream>>>(bw3, sw3, wt3, 2);

    int Npool = B * 169;
    kan_conv1_pool<<<(Npool + 127) / 128, 128, 0, stream>>>(x, bw1, sw1, h1, Npool);

    int Np2 = B * 121;
    kan_conv_wmma<5, 13, 13, 11, 11, 5><<<(Np2 + 31) / 32, 32, 0, stream>>>(h1, wt2, h2, Np2);

    int Np3 = B * 81;
    kan_conv_wmma<5, 11, 11, 9, 9, 2><<<(Np3 + 31) / 32, 32, 0, stream>>>(h2, wt3, h3, Np3);

    int MT = (B + 15) / 16;
    // FC1: N=500 -> NTPW=4, NG = ceil(500/64) = 8
    {
        int NG = 8;
        int waves = MT * NG;
        fc_wmma<4><<<(waves * 32 + 255) / 256, 256, 0, stream>>>(h3, w1, b1, fc1, B, 500, 162, NG, 1);
    }
    // FC2: N=10 -> NTPW=1, NG = 1
    {
        int NG = 1;
        int waves = MT * NG;
        fc_wmma<1><<<(waves * 32 + 255) / 256, 256, 0, stream>>>(fc1, w2, b2, out, B, 10, 500, NG, 0);
    }
}